// SurpriseGatedMemory_16071767622139
// MI455X (gfx1250) — compile-verified
//
#include <hip/hip_runtime.h>

// Chunked WY (delta-rule) reformulation of the surprise-gated fast-weight scan.
// One workgroup per batch; M resident in LDS; GEMMs on v_wmma_f32_16x16x4_f32.
// Round 2: software-pipelined K/V global loads (prefetch next chunk under the
// GEMM phases) and Gram GEMM split across waves 0/1 to balance WMMA work.

typedef float v2f __attribute__((ext_vector_type(2)));
typedef float v8f __attribute__((ext_vector_type(8)));

#define WMMA_F32(a, b, c) \
  __builtin_amdgcn_wmma_f32_16x16x4_f32(false, (a), false, (b), (short)0, (c), false, false)

constexpr int Bn  = 64;    // batch
constexpr int Sn  = 1024;  // sequence
constexpr int DKn = 128;
constexpr int DVn = 128;
constexpr int Cn  = 16;    // chunk (WY block) size
constexpr int NCH = Sn / Cn;
constexpr int LDM = 132;   // padded LDS strides (bank-conflict avoidance)
constexpr int LDU = 132;
constexpr int LDG = 20;

__global__ __launch_bounds__(256, 1)
void sgm_chunked_wy_kernel(const float* __restrict__ mem_in,
                           const float* __restrict__ key,
                           const float* __restrict__ value,
                           float* __restrict__ mem_out,
                           float* __restrict__ gate_out)
{
  __shared__ float Msh[DVn * LDM];    // memory state M[v][k], resident all kernel
  __shared__ float Ush[Cn * LDU];     // normalized keys  u_t[k]
  __shared__ float Vsh[Cn * LDU];     // values           v_t[d]
  __shared__ float Ash[Cn * LDU];     // Ash[t][d] = (M0 @ u_t)[d]
  __shared__ float Wsh[Cn * LDU];     // Wsh[t][d] = -g_t*(1.1 P_t - 0.1 v_t)[d]
  __shared__ float Gsh0[Cn * LDG];    // Gram partial (k = 0..63)
  __shared__ float Gsh1[Cn * LDG];    // Gram partial (k = 64..127)
  __shared__ float part[256];         // norm partials
  __shared__ float invn[Cn];          // 1/(||k_t||+eps)

  const int tid  = threadIdx.x;
  const int lane = tid & 31;
  const int wave = tid >> 5;
  const int b    = blockIdx.x;

  // WMMA lane decomposition (ISA 7.12.2, f32 operands, wave32)
  const int n  = lane & 15;           // C/D column, A row, B column
  const int mh = (lane >> 4) << 3;    // C/D row-half offset (m = vgpr + mh)
  const int kh = (lane >> 4) << 1;    // A/B K-half offset  (k = vgpr + kh)

  // ---- load memory state into LDS ----
  {
    const float* src = mem_in + (size_t)b * DVn * DKn;
    for (int i = tid; i < DVn * DKn; i += 256) {
      const int v = i >> 7, k = i & 127;
      Msh[v * LDM + k] = src[i];
    }
  }

  const int t_loc = tid >> 4;         // 0..15: chunk row handled in stage 1
  const int seg   = tid & 15;         // 0..15: 8-wide segment of the 128 dims
  const size_t kv_base0 = (size_t)b * Sn * DKn + (size_t)t_loc * DKn + seg * 8;

  // ---- prefetch chunk 0 K/V into registers ----
  float kreg[8], vreg[8];
  {
    const float4* kp = (const float4*)(key + kv_base0);
    const float4* vp = (const float4*)(value + kv_base0);
    const float4 k0 = kp[0], k1 = kp[1];
    const float4 v0 = vp[0], v1 = vp[1];
    kreg[0] = k0.x; kreg[1] = k0.y; kreg[2] = k0.z; kreg[3] = k0.w;
    kreg[4] = k1.x; kreg[5] = k1.y; kreg[6] = k1.z; kreg[7] = k1.w;
    vreg[0] = v0.x; vreg[1] = v0.y; vreg[2] = v0.z; vreg[3] = v0.w;
    vreg[4] = v1.x; vreg[5] = v1.y; vreg[6] = v1.z; vreg[7] = v1.w;
  }
  __syncthreads();

  for (int c = 0; c < NCH; ++c) {
    // ======== stage 1: normalize prefetched keys, commit U/V to LDS ========
    {
      float ss = 0.f;
      for (int j = 0; j < 8; ++j) ss += kreg[j] * kreg[j];
      part[tid] = ss;                 // part[t_loc*16 + seg]
    }
    __syncthreads();
    if (tid < Cn) {
      float ss = 0.f;
      for (int s2 = 0; s2 < 16; ++s2) ss += part[tid * 16 + s2];
      invn[tid] = 1.f / (sqrtf(ss) + 1e-6f);
    }
    __syncthreads();
    {
      const float inv = invn[t_loc];
      for (int j = 0; j < 8; ++j) {
        Ush[t_loc * LDU + seg * 8 + j] = kreg[j] * inv;
        Vsh[t_loc * LDU + seg * 8 + j] = vreg[j];
      }
    }
    __syncthreads();

    // ======== software pipeline: issue next chunk's K/V loads now ========
    // (clamped on last chunk; s_wait_loadcnt lands at next iteration's use,
    //  so HBM latency hides under the GEMM phases below)
    {
      const int cnext = (c + 1 < NCH) ? (c + 1) : c;
      const size_t base = kv_base0 + (size_t)(cnext * Cn) * DKn;
      const float4* kp = (const float4*)(key + base);
      const float4* vp = (const float4*)(value + base);
      const float4 k0 = kp[0], k1 = kp[1];
      const float4 v0 = vp[0], v1 = vp[1];
      kreg[0] = k0.x; kreg[1] = k0.y; kreg[2] = k0.z; kreg[3] = k0.w;
      kreg[4] = k1.x; kreg[5] = k1.y; kreg[6] = k1.z; kreg[7] = k1.w;
      vreg[0] = v0.x; vreg[1] = v0.y; vreg[2] = v0.z; vreg[3] = v0.w;
      vreg[4] = v1.x; vreg[5] = v1.y; vreg[6] = v1.z; vreg[7] = v1.w;
    }

    // ======== phase A: Ash = M0 @ U^T  (each wave: one 16-row block) ========
    {
      const int row0 = wave * 16;
      v8f acc; for (int r = 0; r < 8; ++r) acc[r] = 0.f;
      #pragma unroll 8
      for (int k0 = 0; k0 < DKn; k0 += 4) {
        v2f a, bb;
        a.x  = Msh[(row0 + n) * LDM + k0 + kh];       // A[m][kk] = M[row0+m][k]
        a.y  = Msh[(row0 + n) * LDM + k0 + kh + 1];
        bb.x = Ush[n * LDU + k0 + kh];                // B[kk][t] = u_t[k]
        bb.y = Ush[n * LDU + k0 + kh + 1];
        acc = WMMA_F32(a, bb, acc);
      }
      // store transposed: Ash[t][d]  (d = row0 + m, t = n)
      for (int r = 0; r < 8; ++r) Ash[n * LDU + row0 + mh + r] = acc[r];
    }
    // ======== Gram split across waves 0/1: Gp = U[:,ks..ks+63] slice ========
    if (wave < 2) {
      const int kbase = wave * 64;                    // wave0: 0..63, wave1: 64..127
      float* Gp = (wave == 0) ? Gsh0 : Gsh1;
      v8f acc; for (int r = 0; r < 8; ++r) acc[r] = 0.f;
      #pragma unroll 8
      for (int k0 = 0; k0 < 64; k0 += 4) {
        v2f a;                                        // A[i][kk] == B[kk][j] per lane
        a.x = Ush[n * LDU + kbase + k0 + kh];
        a.y = Ush[n * LDU + kbase + k0 + kh + 1];
        acc = WMMA_F32(a, a, acc);
      }
      for (int r = 0; r < 8; ++r) Gp[(mh + r) * LDG + n] = acc[r];
    }
    __syncthreads();

    // ======== sequential tail (wave 0 only; shfl reductions, no barriers) ========
    if (wave == 0) {
      for (int t = 0; t < Cn; ++t) {
        float P[4];
        for (int j = 0; j < 4; ++j)
          P[j] = Ash[t * LDU + lane + 32 * j];        // M0 @ u_t
        for (int i = 0; i < t; ++i) {                 // + in-chunk corrections
          const float g = Gsh0[i * LDG + t] + Gsh1[i * LDG + t];
          for (int j = 0; j < 4; ++j)
            P[j] += Wsh[i * LDU + lane + 32 * j] * g;
        }
        float ss = 0.f;
        float sv[4];
        for (int j = 0; j < 4; ++j) {
          sv[j] = P[j] - Vsh[t * LDU + lane + 32 * j];  // surprise
          ss += sv[j] * sv[j];
        }
        for (int off = 16; off >= 1; off >>= 1)
          ss += __shfl_xor(ss, off, 32);
        const float snorm = sqrtf(ss);
        const float gate = 1.f / (1.f + __expf((0.1f - snorm) * 10.f));
        if (lane == 0)
          gate_out[(size_t)b * Sn + c * Cn + t] = gate;
        for (int j = 0; j < 4; ++j) {
          const int d = lane + 32 * j;
          Wsh[t * LDU + d] = -gate * (1.1f * P[j] - 0.1f * Vsh[t * LDU + d]);
        }
      }
    }
    __syncthreads();

    // ======== update: M += W @ U  (W already carries the -gate sign) ========
    {
      const int row0 = wave * 16;
      for (int jt = 0; jt < 8; ++jt) {
        const int col0 = jt * 16;
        v8f acc;                                      // C = current M tile
        for (int r = 0; r < 8; ++r)
          acc[r] = Msh[(row0 + mh + r) * LDM + col0 + n];
        #pragma unroll
        for (int t0 = 0; t0 < Cn; t0 += 4) {
          v2f a, bb;
          a.x  = Wsh[(t0 + kh) * LDU + row0 + n];     // A[m][tt] = W[row0+m][t]
          a.y  = Wsh[(t0 + kh + 1) * LDU + row0 + n];
          bb.x = Ush[(t0 + kh) * LDU + col0 + n];     // B[tt][nn] = u_t[col0+nn]
          bb.y = Ush[(t0 + kh + 1) * LDU + col0 + n];
          acc = WMMA_F32(a, bb, acc);
        }
        for (int r = 0; r < 8; ++r)
          Msh[(row0 + mh + r) * LDM + col0 + n] = acc[r];
      }
    }
    __syncthreads();
  }

  // ---- write out final memory state ----
  {
    float* dst = mem_out + (size_t)b * DVn * DKn;
    for (int i = tid; i < DVn * DKn; i += 256) {
      const int v = i >> 7, k = i & 127;
      dst[i] = Msh[v * LDM + k];
    }
  }
}

extern "C" void kernel_launch(void* const* d_in, const int* in_sizes, int n_in,
                              void* d_out, int out_size, void* d_ws, size_t ws_size,
                              hipStream_t stream) {
  (void)in_sizes; (void)n_in; (void)d_ws; (void)ws_size; (void)out_size;
  const float* mem = (const float*)d_in[0];   // (B, DV, DK) f32
  const float* key = (const float*)d_in[1];   // (B, S, DK)  f32
  const float* val = (const float*)d_in[2];   // (B, S, DV)  f32
  float* out  = (float*)d_out;
  float* outM = out;                                  // mem_final, B*DV*DK
  float* outG = out + (size_t)Bn * DVn * DKn;         // gates, B*S
  sgm_chunked_wy_kernel<<<dim3(Bn), dim3(256), 0, stream>>>(mem, key, val, outM, outG);
}